// MoNetLayer_7911329759627
// MI455X (gfx1250) — compile-verified
//
#include <hip/hip_runtime.h>
#include <hip/hip_bf16.h>
#include <math.h>

#define WIDTH_IN 64   // DIM + F = 3 + 61
#define DIMC 3
#define RC 3
#define CC 32
#define WIDTH_OUT 35  // DIM + C

typedef __attribute__((ext_vector_type(2))) float v2f;
typedef __attribute__((ext_vector_type(8))) float v8f;

__device__ __forceinline__ float readlane_f(float v, int lane) {
  return __int_as_float(__builtin_amdgcn_readlane(__float_as_int(v), lane));
}

// Branchless tanh: (e^{2x}-1)/(e^{2x}+1), clamped so exp never overflows.
__device__ __forceinline__ float fast_tanh(float x) {
  float xc = fminf(fmaxf(x, -10.f), 10.f);
  float t  = __expf(2.f * xc);
  return (t - 1.f) * __builtin_amdgcn_rcpf(t + 1.f);
}

// Kernel 1: per-node pass. fsum[n] = sum(features[n, 3:64]); out[n,0:3]=coords;
// out[n,3:35]=0 (re-zeroed every call: kernel 3 accumulates atomically).
__global__ void monet_init(const float* __restrict__ feats,
                           float* __restrict__ out,
                           float* __restrict__ fsum,
                           int N) {
  int n = blockIdx.x * blockDim.x + threadIdx.x;
  if (n >= N) return;
  const float4* r4 = (const float4*)(feats + (size_t)n * WIDTH_IN);
  float4 v0 = r4[0];
  float s = v0.w;                       // column 3 is first feature
  #pragma unroll
  for (int k = 1; k < WIDTH_IN / 4; ++k) {
    float4 v = r4[k];
    s += v.x + v.y + v.z + v.w;
  }
  fsum[n] = s;
  float* orow = out + (size_t)n * WIDTH_OUT;
  orow[0] = v0.x; orow[1] = v0.y; orow[2] = v0.z;
  #pragma unroll
  for (int j = DIMC; j < WIDTH_OUT; ++j) orow[j] = 0.f;
}

// Kernel 2: transpose (C,N,3) mu/sigma into packed (N,C,6) = [m0,m1,m2,i0,i1,i2]
// with i = 1/sigma. Makes the per-edge gather 768 contiguous bytes (6 L2 lines
// instead of 64) and hoists the reciprocals out of the hot loop.
__global__ void monet_pack(const float* __restrict__ mu,
                           const float* __restrict__ sigma,
                           float2* __restrict__ pk,
                           int N) {
  long long t = (long long)blockIdx.x * blockDim.x + threadIdx.x;
  if (t >= (long long)N * CC) return;
  const int c = (int)(t & (CC - 1));
  const int n = (int)(t >> 5);
  const float* mp = mu    + ((size_t)c * N + (size_t)n) * RC;
  const float* sp = sigma + ((size_t)c * N + (size_t)n) * RC;
  float2* o = pk + (size_t)t * 3;
  o[0] = make_float2(mp[0], mp[1]);
  o[1] = make_float2(mp[2], __builtin_amdgcn_rcpf(sp[0]));
  o[2] = make_float2(__builtin_amdgcn_rcpf(sp[1]), __builtin_amdgcn_rcpf(sp[2]));
}

// Kernel 3: one wave32 per 16 edges.
// Phase A: V_WMMA_F32_16X16X4_F32 computes du @ W_u (+bias via C) for 16 edges.
// Phase B: loop over the 16 edges; lane = cluster c; readlane broadcasts make
// srcm wave-uniform (saddr-form gathers + one coalesced atomic burst per edge).
template <bool PACKED>
__global__ void monet_edges(const float* __restrict__ feats,
                            const float* __restrict__ adj,
                            const float* __restrict__ mu,
                            const float* __restrict__ sigma,
                            const float* __restrict__ pk,
                            const float* __restrict__ Wu,
                            const float* __restrict__ bu,
                            const int*   __restrict__ edges,
                            const float* __restrict__ fsum,
                            float* __restrict__ out,
                            int N, int E) {
  const int lane = threadIdx.x & 31;
  const int wave = (blockIdx.x * blockDim.x + threadIdx.x) >> 5;
  const long long base = (long long)wave * 16;
  if (base >= E) return;                      // wave-uniform: EXEC stays all-1s

  // ---- Phase A: gather per-edge data (lane m and lane m+16 share edge m) ----
  const int m0 = lane & 15;
  const long long e = base + m0;
  const bool ev = (e < (long long)E);
  int srcL = 0, dstL = 0;
  float avL = 0.f;
  if (ev) {
    srcL = edges[2 * e];
    dstL = edges[2 * e + 1];
    avL  = adj[e];                            // av==0 kills padded-edge contrib
  }
  const float* fs = feats + (size_t)srcL * WIDTH_IN;
  const float* fd = feats + (size_t)dstL * WIDTH_IN;
  float dux = fd[0] - fs[0];
  float duy = fd[1] - fs[1];
  float duz = fd[2] - fs[2];
  if (!ev) { dux = 0.f; duy = 0.f; duz = 0.f; }
  const float fsdL = fsum[dstL];

  // A matrix 16x4 f32: lanes 0-15 -> {K0,K1}, lanes 16-31 -> {K2,K3}; K3 = pad.
  v2f a;
  a.x = (lane < 16) ? dux : duz;
  a.y = (lane < 16) ? duy : 0.f;

  // B matrix 4x16 f32: VGPR0 = rows K0(lanes0-15)/K2(lanes16-31), VGPR1 = K1/K3.
  v2f b; b.x = 0.f; b.y = 0.f;
  if (m0 < RC) {
    if (lane < 16) { b.x = Wu[0 * RC + m0]; b.y = Wu[1 * RC + m0]; }
    else           { b.x = Wu[2 * RC + m0]; }
  }

  // C matrix carries the bias broadcast down each valid column n = m0.
  const float bias = (m0 < RC) ? bu[m0] : 0.f;
  v8f acc;
  #pragma unroll
  for (int i = 0; i < 8; ++i) acc[i] = bias;

  acc = __builtin_amdgcn_wmma_f32_16x16x4_f32(false, a, false, b,
                                              (short)0, acc, false, false);

  // Branchless tanh only in the 6 lanes holding valid columns.
  if (m0 < RC) {
    #pragma unroll
    for (int i = 0; i < 8; ++i) acc[i] = fast_tanh(acc[i]);
  }

  // ---- Phase B: lane = cluster; readlane broadcasts from D layout ----
  #pragma unroll
  for (int m = 0; m < 16; ++m) {
    const int   srcm = __builtin_amdgcn_readlane(srcL, m);
    const float avm  = readlane_f(avL, m);
    const float fsm  = readlane_f(fsdL, m);
    const int   half = (m < 8) ? 0 : 16;      // D[m,n]: VGPR m%8, lane 16*(m/8)+n
    const float u0 = readlane_f(acc[m & 7], half + 0);
    const float u1 = readlane_f(acc[m & 7], half + 1);
    const float u2 = readlane_f(acc[m & 7], half + 2);

    float q;
    if (PACKED) {
      const float2* pp = (const float2*)pk + ((size_t)srcm * CC + lane) * 3;
      const float2 p0 = pp[0], p1 = pp[1], p2 = pp[2];
      const float d0 = u0 - p0.x;
      const float d1 = u1 - p0.y;
      const float d2 = u2 - p1.x;
      q = d0 * d0 * p1.y + d1 * d1 * p2.x + d2 * d2 * p2.y;
    } else {
      const float* mp = mu    + ((size_t)lane * N + (size_t)srcm) * RC;
      const float* sp = sigma + ((size_t)lane * N + (size_t)srcm) * RC;
      const float d0 = u0 - mp[0];
      const float d1 = u1 - mp[1];
      const float d2 = u2 - mp[2];
      q = d0 * d0 * __builtin_amdgcn_rcpf(sp[0])
        + d1 * d1 * __builtin_amdgcn_rcpf(sp[1])
        + d2 * d2 * __builtin_amdgcn_rcpf(sp[2]);
    }
    const float w = __expf(-0.5f * q);
    // 32 lanes -> out[src*35 + 3 .. +34]: one coalesced atomic burst per edge.
    atomicAdd(out + (size_t)srcm * WIDTH_OUT + DIMC + lane, avm * fsm * w);
  }
}

extern "C" void kernel_launch(void* const* d_in, const int* in_sizes, int n_in,
                              void* d_out, int out_size, void* d_ws, size_t ws_size,
                              hipStream_t stream) {
  const float* feats = (const float*)d_in[0];
  const float* adj   = (const float*)d_in[1];
  const float* mu    = (const float*)d_in[2];
  const float* sigma = (const float*)d_in[3];
  const float* Wu    = (const float*)d_in[4];
  const float* bu    = (const float*)d_in[5];
  const int*   edges = (const int*)d_in[6];
  float* out = (float*)d_out;

  const int N = in_sizes[0] / WIDTH_IN;   // features is (N, 64)
  const int E = in_sizes[1];              // adj_vals is (E,)

  const size_t fsumBytes = (((size_t)N * 4) + 255) & ~(size_t)255;
  const size_t pkBytes   = (size_t)N * CC * 6 * sizeof(float);
  const bool packed = (ws_size >= fsumBytes + pkBytes);

  float* fsum = (float*)d_ws;
  float* pk   = (float*)((char*)d_ws + fsumBytes);

  const int b1 = (N + 255) / 256;
  monet_init<<<b1, 256, 0, stream>>>(feats, out, fsum, N);

  const long long waves   = ((long long)E + 15) / 16;
  const long long threads = waves * 32;
  const int b2 = (int)((threads + 255) / 256);

  if (packed) {
    const long long pkThreads = (long long)N * CC;
    const int b3 = (int)((pkThreads + 255) / 256);
    monet_pack<<<b3, 256, 0, stream>>>(mu, sigma, (float2*)pk, N);
    monet_edges<true><<<b2, 256, 0, stream>>>(feats, adj, mu, sigma, pk, Wu, bu,
                                              edges, fsum, out, N, E);
  } else {
    monet_edges<false><<<b2, 256, 0, stream>>>(feats, adj, mu, sigma, nullptr,
                                               Wu, bu, edges, fsum, out, N, E);
  }
}